// ProjLayer_89017492177530
// MI455X (gfx1250) — compile-verified
//
#include <hip/hip_runtime.h>
#include <hip/hip_bf16.h>

// CDNA5 (gfx1250) heatmap generation via separable Gaussian + WMMA outer product.
//
// heat[b,j,r,c] = exp(-((r-ceil(y))^2 + (c-ceil(x))^2)/3)
//               = rowv[r] * colv[c]          (separable -> rank-1 outer product)
//
// One wave32 per (b,j): computes 8 exps/lane, then 16x V_WMMA_F32_16X16X4_F32
// (K=0 carries the vectors, K=1..3 zeroed) to produce the 64x64 tile, streamed
// to HBM with non-temporal stores (704 MB write-once output, HBM-bound).

typedef float v2f __attribute__((ext_vector_type(2)));
typedef float v8f __attribute__((ext_vector_type(8)));

#define N_JOINTS 21
#define HW 64
#define RADIUS 3.0f
#define WAVES_PER_BLOCK 8

__global__ __launch_bounds__(256)
void ProjLayer_89017492177530_kernel(const float* __restrict__ x,
                                     float* __restrict__ out,
                                     int total_bj) {
    const int wave = threadIdx.x >> 5;          // wave id within block (uniform per wave)
    const int lane = threadIdx.x & 31;
    const int bj = blockIdx.x * WAVES_PER_BLOCK + wave;
    if (bj >= total_bj) return;                 // wave-uniform guard (never taken here)

    // Keypoint for this (b, j): x-coordinate then y-coordinate.
    const float px = ceilf(x[(size_t)bj * 2 + 0]);
    const float py = ceilf(x[(size_t)bj * 2 + 1]);
    const float inv_r = 1.0f / RADIUS;

    // Lanes 0..15 supply the K=0 slice of A (rows) and B (cols) for each 16-tile.
    // Lane p (p = lane & 15) precomputes exp terms for positions {p, 16+p, 32+p, 48+p}.
    const int p = lane & 15;
    float rowe[4], cole[4];
#pragma unroll
    for (int t = 0; t < 4; ++t) {
        float dr = (float)(t * 16 + p) - py;
        float dc = (float)(t * 16 + p) - px;
        rowe[t] = __expf(-(dr * dr) * inv_r);
        cole[t] = __expf(-(dc * dc) * inv_r);
    }

    // v_cndmask select (no EXEC divergence): K=0 lives in VGPR0 lanes 0..15 of
    // the 32-bit 16x4 A layout; lanes 16..31 hold K=2 and VGPR1 holds K=1/3 -> 0.
    const float keep = (lane < 16) ? 1.0f : 0.0f;

    float* outp = out + (size_t)bj * (HW * HW);
    const int n = lane & 15;                    // column within 16x16 tile
    const int mhalf = (lane >> 4) * 8;          // D layout: lanes 16..31 hold M+8

#pragma unroll
    for (int tr = 0; tr < 4; ++tr) {
        v2f A;
        A.x = rowe[tr] * keep;                  // A[m, K=0] = rowv[tr*16 + m]
        A.y = 0.0f;                             // K=1 (lanes<16) / K=3 (lanes>=16)
#pragma unroll
        for (int tc = 0; tc < 4; ++tc) {
            v2f B;
            B.x = cole[tc] * keep;              // B[K=0, n] = colv[tc*16 + n]
            B.y = 0.0f;
            v8f C = {0.0f, 0.0f, 0.0f, 0.0f, 0.0f, 0.0f, 0.0f, 0.0f};
            // D = A x B + C  (rank-1 outer product; K=1..3 contribute zero)
            v8f D = __builtin_amdgcn_wmma_f32_16x16x4_f32(
                false, A, false, B, (short)0, C, false, false);

            // D VGPR i, lane l: element (tr*16 + i + 8*(l>>4), tc*16 + (l&15)).
            float* base = outp + (size_t)(tr * 16 + mhalf) * HW + tc * 16 + n;
#pragma unroll
            for (int i = 0; i < 8; ++i) {
                __builtin_nontemporal_store(D[i], base + (size_t)i * HW);
            }
        }
    }
}

extern "C" void kernel_launch(void* const* d_in, const int* in_sizes, int n_in,
                              void* d_out, int out_size, void* d_ws, size_t ws_size,
                              hipStream_t stream) {
    const float* x = (const float*)d_in[0];
    float* out = (float*)d_out;

    const int total_bj = in_sizes[0] / 2;       // 2048 * 21 = 43008 keypoints
    const int blocks = (total_bj + WAVES_PER_BLOCK - 1) / WAVES_PER_BLOCK;  // 5376

    ProjLayer_89017492177530_kernel<<<blocks, 256, 0, stream>>>(x, out, total_bj);
}